// QRNNEncoder_58712202936919
// MI455X (gfx1250) — compile-verified
//
#include <hip/hip_runtime.h>
#include <math.h>

// QRNN encoder: T=512, B=32, E=512, H=2048
// Phase 1: Y = sent(T*B,E) x W^T(E,3H) + b, fused tanh/sigmoid -> ws (T*B, 3H) fp32
//          (WMMA f32 16x16x4, async-to-LDS double-buffered pipeline)
// Phase 2: sequential forget-mult scan over T + running max -> out (B, H)

typedef __attribute__((ext_vector_type(2))) float v2f;
typedef __attribute__((ext_vector_type(8))) float v8f;

#define T_DIM 512
#define B_DIM 32
#define E_DIM 512
#define H_DIM 2048
#define M_TOT (T_DIM * B_DIM)   // 16384
#define N_TOT (3 * H_DIM)       // 6144
#define K_TOT E_DIM             // 512

#define BM 128
#define BN 128
#define BK 32
#define LDA 36                  // A row stride: rows 16B-aligned for async b128,
                                // conflict-free b64 fragment reads
#define BNP (2 * BN + 32)       // B pair-row stride: row p vs p+1 differ by 32 banks

// ---- CDNA5 async global->LDS (ASYNCcnt) ----
__device__ __forceinline__ void async_ld_b128(unsigned lds_off, const float* g) {
    asm volatile("global_load_async_to_lds_b128 %0, %1, off"
                 :: "v"(lds_off), "v"(g) : "memory");
}
__device__ __forceinline__ void async_ld_b64(unsigned lds_off, const float* g) {
    asm volatile("global_load_async_to_lds_b64 %0, %1, off"
                 :: "v"(lds_off), "v"(g) : "memory");
}
__device__ __forceinline__ void wait_async0() {
    asm volatile("s_wait_asynccnt 0x0" ::: "memory");
}

__global__ __launch_bounds__(256) void qrnn_gemm_act_kernel(
    const float* __restrict__ sent,   // (M_TOT, K_TOT) row-major (= (T,B,E))
    const float* __restrict__ W,      // (N_TOT, K_TOT) row-major
    const float* __restrict__ bias,   // (N_TOT)
    float* __restrict__ ws)           // (M_TOT, N_TOT) activated gates
{
    __shared__ float As[2][BM * LDA];          // A: (m, k) padded, double-buffered
    __shared__ float Bs[2][(BK / 2) * BNP];    // B: (k/2, n, 2) pair-interleaved

    const int tid  = threadIdx.x;
    const int m0   = blockIdx.x * BM;
    const int n0   = blockIdx.y * BN;
    const int wave = tid >> 5;
    const int lane = tid & 31;
    const int half = lane >> 4;       // 0: lanes 0-15, 1: lanes 16-31
    const int l16  = lane & 15;
    const int wm   = wave & 3;        // wave m-offset: 32*wm
    const int wn   = wave >> 2;       // wave n-offset: 64*wn

    const unsigned asB[2] = {(unsigned)(uintptr_t)&As[0][0],
                             (unsigned)(uintptr_t)&As[1][0]};
    const unsigned bsB[2] = {(unsigned)(uintptr_t)&Bs[0][0],
                             (unsigned)(uintptr_t)&Bs[1][0]};

    // Issue one K-chunk's tile loads (async, no VGPR staging).
    auto issue_chunk = [&](int k0, int buf) {
#pragma unroll
        for (int i = 0; i < 4; ++i) {            // A: 128x32 as 1024 b128
            int linear = tid + 256 * i;
            int m  = linear >> 3;
            int k4 = (linear & 7) << 2;
            async_ld_b128(asB[buf] + (unsigned)((m * LDA + k4) * 4),
                          sent + (size_t)(m0 + m) * K_TOT + k0 + k4);
        }
#pragma unroll
        for (int i = 0; i < 8; ++i) {            // B: transpose-scatter as 2048 b64
            int linear = tid + 256 * i;
            int pp = linear & 15;                // K-pair index 0..15
            int n  = linear >> 4;                // 0..127
            async_ld_b64(bsB[buf] + (unsigned)((pp * BNP + 2 * n) * 4),
                         W + (size_t)(n0 + n) * K_TOT + k0 + 2 * pp);
        }
    };

    v8f acc[2][4];
#pragma unroll
    for (int mi = 0; mi < 2; ++mi)
#pragma unroll
        for (int ni = 0; ni < 4; ++ni)
            acc[mi][ni] = (v8f){0.f, 0.f, 0.f, 0.f, 0.f, 0.f, 0.f, 0.f};

    issue_chunk(0, 0);
    wait_async0();
    __syncthreads();

    int cur = 0;
    for (int k0 = 0; k0 < K_TOT; k0 += BK) {
        if (k0 + BK < K_TOT)
            issue_chunk(k0 + BK, cur ^ 1);       // prefetch next chunk

        const float* Ac = &As[cur][0];
        const float* Bc = &Bs[cur][0];
        // ---- 8 K-steps of 4, 2x4 WMMA tiles per wave ----
#pragma unroll
        for (int s = 0; s < 8; ++s) {
            const int kk = 4 * s + 2 * half;  // lanes 0-15: K 0,1 ; 16-31: K 2,3
            const int p  = kk >> 1;           // = 2*s + half
            v2f a[2], b[4];
#pragma unroll
            for (int mi = 0; mi < 2; ++mi)
                a[mi] = *(const v2f*)(&Ac[(32 * wm + 16 * mi + l16) * LDA + kk]);
#pragma unroll
            for (int ni = 0; ni < 4; ++ni) {
                const int nc = 64 * wn + 16 * ni + l16;
                b[ni] = *(const v2f*)(&Bc[p * BNP + 2 * nc]);
            }
#pragma unroll
            for (int mi = 0; mi < 2; ++mi)
#pragma unroll
                for (int ni = 0; ni < 4; ++ni)
                    acc[mi][ni] = __builtin_amdgcn_wmma_f32_16x16x4_f32(
                        false, a[mi], false, b[ni], (short)0, acc[mi][ni],
                        false, false);
        }

        wait_async0();                           // prefetched tiles have landed
        __syncthreads();
        cur ^= 1;
    }

    // ---- epilogue: bias + activation (block is entirely in one gate) ----
    const int gate = n0 / H_DIM;      // 0 = Z (tanh), 1/2 = F/O (sigmoid)
#pragma unroll
    for (int mi = 0; mi < 2; ++mi) {
#pragma unroll
        for (int ni = 0; ni < 4; ++ni) {
            const int ncol = n0 + 64 * wn + 16 * ni + l16;
            const float bv = bias[ncol];
#pragma unroll
            for (int j = 0; j < 8; ++j) {
                const int mrow = m0 + 32 * wm + 16 * mi + 8 * half + j;
                float y = acc[mi][ni][j] + bv;
                float r = (gate == 0) ? tanhf(y)
                                      : 1.0f / (1.0f + __expf(-y));
                ws[(size_t)mrow * N_TOT + ncol] = r;
            }
        }
    }
}

__global__ __launch_bounds__(256) void qrnn_scan_max_kernel(
    const float* __restrict__ ws,     // (T*B, 3H) activated gates
    float* __restrict__ out)          // (B, H)
{
    const int tid = blockIdx.x * 256 + threadIdx.x;   // 0..65535
    const int b = tid >> 11;          // / H_DIM
    const int h = tid & (H_DIM - 1);

    float c = 0.0f;
    float best = -INFINITY;
    for (int t = 0; t < T_DIM; ++t) {
        const float* row = ws + (size_t)(t * B_DIM + b) * N_TOT + h;
        const float z = row[0];
        const float f = row[H_DIM];
        const float o = row[2 * H_DIM];
        c = fmaf(f, z - c, c);        // c = f*z + (1-f)*c
        best = fmaxf(best, o * c);
    }
    out[(size_t)b * H_DIM + h] = best;
}

extern "C" void kernel_launch(void* const* d_in, const int* in_sizes, int n_in,
                              void* d_out, int out_size, void* d_ws, size_t ws_size,
                              hipStream_t stream) {
    const float* sent = (const float*)d_in[0];   // (T, B, E)
    // d_in[1] = lengths (unused by the math)
    const float* W    = (const float*)d_in[2];   // (3H, E)
    const float* bias = (const float*)d_in[3];   // (3H)
    float* out = (float*)d_out;                  // (B, H)
    float* ws  = (float*)d_ws;                   // needs M_TOT*N_TOT*4 = 384 MiB

    dim3 grid1(M_TOT / BM, N_TOT / BN);          // 128 x 48 blocks
    qrnn_gemm_act_kernel<<<grid1, 256, 0, stream>>>(sent, W, bias, ws);

    dim3 grid2((B_DIM * H_DIM) / 256);           // 256 blocks
    qrnn_scan_max_kernel<<<grid2, 256, 0, stream>>>(ws, out);
}